// LSTMLayer_63539746177360
// MI455X (gfx1250) — compile-verified
//
#include <hip/hip_runtime.h>

// CDNA5 (gfx1250) fused LSTM: one wave owns 16 batch rows for all 2048 steps.
// gates (64 x 16batch) = [w_ih | w_hh] x [x ; h]^T via V_WMMA_F32_16X16X4_F32.
// C/D layout puts batch in N (lane%16); w_hh's K dimension is permuted (kappa)
// so the C-layout h registers ARE the hh B operand: hB[c] = {h[2c],h[2c+1]} on
// every lane -> zero shuffles on the recurrent critical path.
// The x-projection (bias + w_ih*x) for step t+1 is computed at the end of
// step t (explicit software pipeline); the hh GEMM then accumulates directly
// onto it via the WMMA C operand -> no separate merge adds.

typedef __attribute__((ext_vector_type(2))) float v2f;
typedef __attribute__((ext_vector_type(8))) float v8f;

#define LOG2E 1.44269504088896340736f

__device__ __forceinline__ float fast_sigmoid(float x) {
    // 1 / (1 + 2^(-x*log2e))
    float e = __builtin_amdgcn_exp2f(-x * LOG2E);
    return __builtin_amdgcn_rcpf(1.0f + e);
}

__device__ __forceinline__ float fast_tanh(float x) {
    // 1 - 2/(2^(2x*log2e) + 1); saturates correctly at +-1
    float e = __builtin_amdgcn_exp2f(x * (2.0f * LOG2E));
    return 1.0f - 2.0f * __builtin_amdgcn_rcpf(e + 1.0f);
}

__global__ __launch_bounds__(32) void lstm_fused_kernel(
    const float* __restrict__ images,   // [32, 1024, 64, 16]
    const float* __restrict__ w_ih,     // [64, 16]
    const float* __restrict__ w_hh,     // [64, 16]
    const float* __restrict__ b_ih,     // [64]
    const float* __restrict__ b_hh,     // [64]
    float* __restrict__ out)            // [32, 1024, 64, 16]
{
    const int lane = threadIdx.x;       // 0..31 (wave32)
    const int ln   = lane & 15;         // batch column / A-row select
    const int lh   = lane >> 4;         // lane half
    const int bcol = blockIdx.x * 16 + ln;  // this lane's batch column

    // ---- Preload A-matrices (weights), VGPR-resident for all 2048 steps.
    // A layout (16x4 f32): lane holds M=lane%16, VGPR j holds K = 2*(lane/16)+j.
    //
    // w_ih: natural K ordering; chunk c covers d = 4c + 2*lh + {0,1}.
    // w_hh: kappa-permuted K ordering so the recurrent B operand needs no
    //       lane permute: K-slot (c,lh,j) holds column hid = 2c + 8*lh + j.
    v2f wih[4][4], whh[4][4];
#pragma unroll
    for (int gt = 0; gt < 4; ++gt)
#pragma unroll
        for (int c = 0; c < 4; ++c) {
            int row = (16 * gt + ln) * 16;
            wih[gt][c] = *(const v2f*)(w_ih + row + 4 * c + 2 * lh);
            whh[gt][c] = *(const v2f*)(w_hh + row + 2 * c + 8 * lh);
        }

    // Bias seed in C layout: VGPR r, lane -> gate = 16*gt + r + 8*lh.
    v8f biasC[4];
#pragma unroll
    for (int gt = 0; gt < 4; ++gt)
#pragma unroll
        for (int r = 0; r < 8; ++r) {
            int g = 16 * gt + 8 * lh + r;
            biasC[gt][r] = b_ih[g] + b_hh[g];
        }

    // Recurrent state, C layout: lane -> batch = bcol, VGPR r -> hidden = r + 8*lh.
    v8f h  = {};
    v8f cs = {};

    // ---- Pipeline prologue: acc[gt] = bias + w_ih * x(t=0).
    v8f acc[4];
    {
        const int base0 = (bcol * 64) * 16;   // n=0, s=0
        v2f xB[4];
#pragma unroll
        for (int c = 0; c < 4; ++c)
            xB[c] = *(const v2f*)(images + base0 + 4 * c + 2 * lh);
#pragma unroll
        for (int gt = 0; gt < 4; ++gt) {
            v8f a = biasC[gt];
#pragma unroll
            for (int c = 0; c < 4; ++c)
                a = __builtin_amdgcn_wmma_f32_16x16x4_f32(
                        false, wih[gt][c], false, xB[c], (short)0, a, false, false);
            acc[gt] = a;
        }
    }

    for (int t = 0; t < 2048; ++t) {
        const int n = t >> 6;
        const int s = t & 63;
        const int base = ((n * 1024 + bcol) * 64 + s) * 16;

        // ---- Issue next step's x loads early (consumed at loop bottom).
        const int tp = t < 2047 ? t + 1 : 2047;
        const int np = tp >> 6, sp = tp & 63;
        const int basep = ((np * 1024 + bcol) * 64 + sp) * 16;
        v2f xB[4];
#pragma unroll
        for (int c = 0; c < 4; ++c)
            xB[c] = *(const v2f*)(images + basep + 4 * c + 2 * lh);

        // Prefetch further ahead (streams images through caches).
        {
            int tq = t + 8 < 2048 ? t + 8 : 2047;
            int nq = tq >> 6, sq = tq & 63;
            const float* p = images + ((nq * 1024 + bcol) * 64 + sq) * 16 + 2 * lh;
            __builtin_prefetch(p, 0, 3);
        }

        // ---- hh GEMM accumulates directly onto acc (= bias + x-proj of t).
        // Critical path from h: 4 chained WMMAs, no shuffles, no merge adds.
        v2f hB[4];
#pragma unroll
        for (int c = 0; c < 4; ++c)
            hB[c] = (v2f){h[2 * c], h[2 * c + 1]};
#pragma unroll
        for (int gt = 0; gt < 4; ++gt) {
            v8f a = acc[gt];
#pragma unroll
            for (int c = 0; c < 4; ++c)
                a = __builtin_amdgcn_wmma_f32_16x16x4_f32(
                        false, whh[gt][c], false, hB[c], (short)0, a, false, false);
            acc[gt] = a;
        }

        // ---- LSTM cell (elementwise, co-executes with the XDL pipe).
#pragma unroll
        for (int r = 0; r < 8; ++r) {
            float iv = fast_sigmoid(acc[0][r]);
            float fv = fast_sigmoid(acc[1][r]);
            float gv = fast_tanh   (acc[2][r]);
            float ov = fast_sigmoid(acc[3][r]);
            float cn = fv * cs[r] + iv * gv;
            cs[r] = cn;
            h[r]  = ov * fast_tanh(cn);
        }

        // ---- Store h: out[n, bcol, s, hidden], hidden = 8*lh + r (contiguous).
        float4 o0 = {h[0], h[1], h[2], h[3]};
        float4 o1 = {h[4], h[5], h[6], h[7]};
        *(float4*)(out + base + 8 * lh)     = o0;
        *(float4*)(out + base + 8 * lh + 4) = o1;

        // ---- Pipeline: x-projection for step t+1 (independent of h/cs).
#pragma unroll
        for (int gt = 0; gt < 4; ++gt) {
            v8f a = biasC[gt];
#pragma unroll
            for (int c = 0; c < 4; ++c)
                a = __builtin_amdgcn_wmma_f32_16x16x4_f32(
                        false, wih[gt][c], false, xB[c], (short)0, a, false, false);
            acc[gt] = a;
        }
    }
}

extern "C" void kernel_launch(void* const* d_in, const int* in_sizes, int n_in,
                              void* d_out, int out_size, void* d_ws, size_t ws_size,
                              hipStream_t stream) {
    (void)in_sizes; (void)n_in; (void)out_size; (void)d_ws; (void)ws_size;
    const float* images = (const float*)d_in[0];
    const float* w_ih   = (const float*)d_in[1];
    const float* w_hh   = (const float*)d_in[2];
    const float* b_ih   = (const float*)d_in[3];
    const float* b_hh   = (const float*)d_in[4];
    float* out = (float*)d_out;

    // 1024 batch rows / 16 per wave = 64 one-wave workgroups.
    lstm_fused_kernel<<<dim3(64), dim3(32), 0, stream>>>(
        images, w_ih, w_hh, b_ih, b_hh, out);
}